// DLT_17136919511243
// MI455X (gfx1250) — compile-verified
//
#include <hip/hip_runtime.h>

// ---------------------------------------------------------------------------
// Batched 8x8 DLT solve, one batch element per thread (wave32-safe).
//
// MI455X reasoning: 50 MB total traffic -> ~2.2us memory floor @ 23.3 TB/s;
// ~1k VALU instr/thread of pivoted elimination dominates. WMMA (16x16 dense
// MMA, EXEC all-ones) cannot express the data-dependent partial pivoting this
// solve requires (A[1][1]==0 structurally, pivoting is mandatory), so the
// CDNA5 features used are the async global->LDS path (ASYNCcnt +
// s_wait_asynccnt) for input staging and LDS-staged fully coalesced output.
// ---------------------------------------------------------------------------

#define LDS_AS __attribute__((address_space(3)))
#define GLB_AS __attribute__((address_space(1)))

typedef int v4i __attribute__((vector_size(16)));

#if defined(__gfx1250__) && __has_builtin(__builtin_amdgcn_global_load_async_to_lds_b128)
#define USE_ASYNC_LDS 1
#else
#define USE_ASYNC_LDS 0
#endif

__launch_bounds__(256)
__global__ void dlt_solve_kernel(const float* __restrict__ src,
                                 const float* __restrict__ dst,
                                 float* __restrict__ out)
{
    // smem: input staging 2048 (src) + 2048 (dst) floats = 16 KB,
    // later reused as 2304-float output staging buffer.
    __shared__ float smem[4096];

    const int tid = threadIdx.x;
    const int blk = blockIdx.x;

    float sx[8];  // 4 src points (x,y interleaved)
    float dx[8];  // 4 dst points (u,v interleaved)

#if USE_ASYNC_LDS
    {
        const float* sbase = src + (size_t)blk * 2048;
        const float* dbase = dst + (size_t)blk * 2048;
        // 4 async b128 transfers: 16B per lane per instruction (ASYNCcnt).
        __builtin_amdgcn_global_load_async_to_lds_b128(
            (GLB_AS v4i*)(sbase + tid * 4),        (LDS_AS v4i*)&smem[tid * 4],        0, 0);
        __builtin_amdgcn_global_load_async_to_lds_b128(
            (GLB_AS v4i*)(sbase + 1024 + tid * 4), (LDS_AS v4i*)&smem[1024 + tid * 4], 0, 0);
        __builtin_amdgcn_global_load_async_to_lds_b128(
            (GLB_AS v4i*)(dbase + tid * 4),        (LDS_AS v4i*)&smem[2048 + tid * 4], 0, 0);
        __builtin_amdgcn_global_load_async_to_lds_b128(
            (GLB_AS v4i*)(dbase + 1024 + tid * 4), (LDS_AS v4i*)&smem[3072 + tid * 4], 0, 0);
        asm volatile("s_wait_asynccnt 0x0" ::: "memory");
        __syncthreads();
        const float4 s0 = *(const float4*)&smem[tid * 8];
        const float4 s1 = *(const float4*)&smem[tid * 8 + 4];
        const float4 d0 = *(const float4*)&smem[2048 + tid * 8];
        const float4 d1 = *(const float4*)&smem[2048 + tid * 8 + 4];
        sx[0]=s0.x; sx[1]=s0.y; sx[2]=s0.z; sx[3]=s0.w;
        sx[4]=s1.x; sx[5]=s1.y; sx[6]=s1.z; sx[7]=s1.w;
        dx[0]=d0.x; dx[1]=d0.y; dx[2]=d0.z; dx[3]=d0.w;
        dx[4]=d1.x; dx[5]=d1.y; dx[6]=d1.z; dx[7]=d1.w;
    }
#else
    {
        const size_t g = (size_t)blk * 256 + tid;
        const float4* sp = (const float4*)(src + g * 8);
        const float4* dp = (const float4*)(dst + g * 8);
        const float4 s0 = sp[0], s1 = sp[1];
        const float4 d0 = dp[0], d1 = dp[1];
        sx[0]=s0.x; sx[1]=s0.y; sx[2]=s0.z; sx[3]=s0.w;
        sx[4]=s1.x; sx[5]=s1.y; sx[6]=s1.z; sx[7]=s1.w;
        dx[0]=d0.x; dx[1]=d0.y; dx[2]=d0.z; dx[3]=d0.w;
        dx[4]=d1.x; dx[5]=d1.y; dx[6]=d1.z; dx[7]=d1.w;
    }
#endif

    // Build augmented system M = [A | b], 8x9, fully in registers.
    // Row 2i   : [x, y, 1, 0, 0, 0, -u*x, -u*y | u]
    // Row 2i+1 : [0, 0, 0, x, y, 1, -v*x, -v*y | v]
    float M[8][9];
#pragma unroll
    for (int i = 0; i < 4; ++i) {
        const float x = sx[2 * i], y = sx[2 * i + 1];
        const float u = dx[2 * i], v = dx[2 * i + 1];
        float* r0 = M[2 * i];
        float* r1 = M[2 * i + 1];
        r0[0] = x;    r0[1] = y;    r0[2] = 1.0f;
        r0[3] = 0.0f; r0[4] = 0.0f; r0[5] = 0.0f;
        r0[6] = -u * x; r0[7] = -u * y; r0[8] = u;
        r1[0] = 0.0f; r1[1] = 0.0f; r1[2] = 0.0f;
        r1[3] = x;    r1[4] = y;    r1[5] = 1.0f;
        r1[6] = -v * x; r1[7] = -v * y; r1[8] = v;
    }

    // Gaussian elimination with partial pivoting. Pivoting done via
    // branchless compare-swap bubbling so every array index stays constant
    // after unrolling (keeps M entirely in VGPRs, no scratch).
#pragma unroll
    for (int k = 0; k < 8; ++k) {
#pragma unroll
        for (int r = k + 1; r < 8; ++r) {
            const bool sw = __builtin_fabsf(M[r][k]) > __builtin_fabsf(M[k][k]);
#pragma unroll
            for (int c = k; c < 9; ++c) {
                const float a = M[k][c];
                const float b = M[r][c];
                M[k][c] = sw ? b : a;
                M[r][c] = sw ? a : b;
            }
        }
        const float inv = 1.0f / M[k][k];
#pragma unroll
        for (int r = k + 1; r < 8; ++r) {
            const float f = M[r][k] * inv;
#pragma unroll
            for (int c = k + 1; c < 9; ++c) {
                M[r][c] = __builtin_fmaf(-f, M[k][c], M[r][c]);
            }
        }
    }

    // Back substitution (upper-triangular).
    float h[8];
#pragma unroll
    for (int r = 7; r >= 0; --r) {
        float s = M[r][8];
#pragma unroll
        for (int c = r + 1; c < 8; ++c) {
            s = __builtin_fmaf(-M[r][c], h[c], s);
        }
        h[r] = s / M[r][r];
    }

    // Stage outputs in LDS (stride 9 floats: odd stride -> conflict-free on
    // 64 banks), then stream out with unit-stride fully coalesced stores.
    __syncthreads();  // all input reads of smem done before overwrite
#pragma unroll
    for (int j = 0; j < 8; ++j) smem[tid * 9 + j] = h[j];
    smem[tid * 9 + 8] = 1.0f;
    __syncthreads();

    float* ob = out + (size_t)blk * 2304;
#pragma unroll
    for (int j = 0; j < 9; ++j) {
        const int idx = j * 256 + tid;
        ob[idx] = smem[idx];
    }
}

extern "C" void kernel_launch(void* const* d_in, const int* in_sizes, int n_in,
                              void* d_out, int out_size, void* d_ws, size_t ws_size,
                              hipStream_t stream) {
    const float* src = (const float*)d_in[0];   // [B, 4, 2] float32
    const float* dst = (const float*)d_in[1];   // [B, 4, 2] float32
    float* out = (float*)d_out;                 // [B, 3, 3] float32

    const int B = in_sizes[0] / 8;              // B*N*2 / 8
    const int blocks = B / 256;                 // B = 524288 -> 2048 blocks

    dlt_solve_kernel<<<dim3(blocks), dim3(256), 0, stream>>>(src, dst, out);
}